// DynamicMessagePassing_31190052504237
// MI455X (gfx1250) — compile-verified
//
#include <hip/hip_runtime.h>
#include <stdint.h>

// ---------------------------------------------------------------- constants
#define B_          32
#define H_          256
#define W_          256
#define HW_         (H_*W_)
#define NPIX        (B_*HW_)         // 2,097,152 cells per channel
#define STEPS_      16
#define DECAY_      0.98f
#define DELAY_STEP_ 8
#define MAXN_       2.0f
#define THRESH_     0.35f

// reduction partial slots (spread atomics over 64 L2 addresses per stat)
#define PSLOTS      64
#define ACC_NORM0   0
#define ACC_STEP(s) (PSLOTS + ((s)-1)*3*PSLOTS)   // [pw | diff | norm] x PSLOTS
#define ACC_SAT     (PSLOTS + STEPS_*3*PSLOTS)
#define ACC_LEAK    (ACC_SAT + PSLOTS)
#define ACC_TOTAL   (ACC_LEAK + PSLOTS)

typedef float v2f __attribute__((ext_vector_type(2)));
typedef float v8f __attribute__((ext_vector_type(8)));

// ---------------------------------------------------------------- helpers
// wave32 shuffle reduce -> 8 wave partials in LDS -> total in tid 0
__device__ __forceinline__ float block_reduce(float v, volatile float* sbuf, int tid) {
  for (int off = 16; off > 0; off >>= 1) v += __shfl_down(v, off, 32);
  if ((tid & 31) == 0) sbuf[tid >> 5] = v;
  __syncthreads();
  float r = (tid < 8) ? sbuf[tid] : 0.f;
  for (int off = 4; off > 0; off >>= 1) r += __shfl_down(r, off, 32);
  __syncthreads();
  return r;   // valid at tid==0
}

// CDNA5 async global->LDS copy (ASYNCcnt); per-lane gather into LDS
__device__ __forceinline__ void async_g2l_b32(uint32_t lds_addr, const float* gptr) {
  asm volatile("global_load_async_to_lds_b32 %0, %1, off"
               :: "v"(lds_addr), "v"(gptr) : "memory");
}
__device__ __forceinline__ void wait_asynccnt0() {
  asm volatile("s_wait_asynccnt 0" ::: "memory");
}

// ---------------------------------------------------------------- init
// state = src * free ; flow = state ; accumulate norms[step 0]
__global__ __launch_bounds__(256) void wave_init(
    const float* __restrict__ x,
    float* __restrict__ sr, float* __restrict__ si,
    float* __restrict__ fr, float* __restrict__ fi,
    float* __restrict__ acc) {
  __shared__ float sbuf[8];
  const int lx = threadIdx.x, ly = threadIdx.y;
  const int tid = ly*16 + lx;
  const int b = blockIdx.z;
  const int yy = blockIdx.y*16 + ly, xx = blockIdx.x*16 + lx;
  const int pix = yy*W_ + xx;
  const size_t xb = (size_t)b*5*HW_;
  const size_t sidx = (size_t)b*HW_ + pix;

  float wall = x[xb + pix];
  float fre  = 1.f - wall;
  float r = x[xb + 1*HW_ + pix] * fre;
  float i = x[xb + 2*HW_ + pix] * fre;
  sr[sidx] = r;  si[sidx] = i;
  fr[sidx] = r;  fi[sidx] = i;

  float tot = block_reduce(sqrtf(r*r + i*i), sbuf, tid);
  if (tid == 0) {
    int slot = (blockIdx.x + (blockIdx.y << 4) + (blockIdx.z << 8)) & (PSLOTS-1);
    atomicAdd(&acc[ACC_NORM0 + slot], tot);
  }
}

// ---------------------------------------------------------------- one step
// 16x16 tile + 1-halo staged into LDS via async loads; stencil, clamp,
// decay-update, per-step reductions. State updates in place (pointwise).
__global__ __launch_bounds__(256) void wave_step(
    const float* __restrict__ x,
    const float* __restrict__ fin_r, const float* __restrict__ fin_i,
    float* __restrict__ fout_r, float* __restrict__ fout_i,
    float* __restrict__ sr, float* __restrict__ si,
    float* __restrict__ acc, int step) {
  __shared__ float tr[18][20];
  __shared__ float ti[18][20];
  __shared__ float sbuf[8];
  const int lx = threadIdx.x, ly = threadIdx.y;
  const int tid = ly*16 + lx;
  const int b = blockIdx.z;
  const int y0 = blockIdx.y*16, x0 = blockIdx.x*16;
  const float* br = fin_r + (size_t)b*HW_;
  const float* bi = fin_i + (size_t)b*HW_;

  // stage 18x18 halo tile: async global->LDS for in-bounds, zero for pad
  for (int l = tid; l < 18*18; l += 256) {
    int hy = l / 18, hx = l % 18;
    int gy = y0 + hy - 1, gx = x0 + hx - 1;
    if ((unsigned)gy < (unsigned)H_ && (unsigned)gx < (unsigned)W_) {
      uint32_t ar = (uint32_t)(uintptr_t)&tr[hy][hx];
      uint32_t ai = (uint32_t)(uintptr_t)&ti[hy][hx];
      async_g2l_b32(ar, br + gy*W_ + gx);
      async_g2l_b32(ai, bi + gy*W_ + gx);
    } else {
      tr[hy][hx] = 0.f;
      ti[hy][hx] = 0.f;
    }
  }
  wait_asynccnt0();
  __syncthreads();

  // 4-neighbor stencil (zero padded)
  float ir = tr[ly][lx+1] + tr[ly+2][lx+1] + tr[ly+1][lx] + tr[ly+1][lx+2];
  float ii = ti[ly][lx+1] + ti[ly+2][lx+1] + ti[ly+1][lx] + ti[ly+1][lx+2];

  const int yy = y0 + ly, xx = x0 + lx;
  const int pix = yy*W_ + xx;
  const size_t xb = (size_t)b*5*HW_;
  float wall = x[xb + pix];
  float fre  = 1.f - wall;
  if (step == DELAY_STEP_) {           // uniform (scalar) branch
    ir += x[xb + 3*HW_ + pix];
    ii += x[xb + 4*HW_ + pix];
  }
  ir *= fre;  ii *= fre;
  float mag = sqrtf(ir*ir + ii*ii + 1e-8f);
  float sc  = fminf(MAXN_ / mag, 1.f);
  ir *= sc;  ii *= sc;
  float pw = (fabsf(ir) + fabsf(ii)) * wall;

  const size_t sidx = (size_t)b*HW_ + pix;
  float osr = sr[sidx], osi = si[sidx];
  float nr = DECAY_*osr + ir, ni = DECAY_*osi + ii;
  float m2  = sqrtf(nr*nr + ni*ni + 1e-8f);
  float sc2 = fminf(MAXN_ / m2, 1.f);
  nr = nr*sc2*fre;  ni = ni*sc2*fre;

  float dr = nr - osr, di = ni - osi;
  float dif = dr*dr + di*di;
  float nrm = sqrtf(nr*nr + ni*ni);

  fout_r[sidx] = ir;  fout_i[sidx] = ii;
  sr[sidx] = nr;      si[sidx] = ni;

  float pwT  = block_reduce(pw,  sbuf, tid);
  float difT = block_reduce(dif, sbuf, tid);
  float nrmT = block_reduce(nrm, sbuf, tid);

  const int slot = (blockIdx.x + (blockIdx.y << 4) + (blockIdx.z << 8)) & (PSLOTS-1);
  float* as = acc + ACC_STEP(step);
  if (tid == 0) {
    atomicAdd(&as[slot],            pwT);
    atomicAdd(&as[PSLOTS   + slot], difT);
    atomicAdd(&as[2*PSLOTS + slot], nrmT);
  }
  if (step == STEPS_) {                // final-state stats fused into last step
    float sat = (fabsf(nr) > 4.f ? 1.f : 0.f) + (fabsf(ni) > 4.f ? 1.f : 0.f);
    float lk  = (nrm > 0.5f && wall > 0.5f) ? 1.f : 0.f;
    float satT = block_reduce(sat, sbuf, tid);
    float lkT  = block_reduce(lk,  sbuf, tid);
    if (tid == 0) {
      atomicAdd(&acc[ACC_SAT  + slot], satT);
      atomicAdd(&acc[ACC_LEAK + slot], lkT);
    }
  }
}

// ---------------------------------------------------------------- finalize
// single wave32 block: readout matmul via v_wmma_f32_16x16x4_f32 (two 16-row
// tiles, K padded 2->4, N padded 8->16) + scalar stats. EXEC is all-ones up
// through the WMMAs (ISA requirement).
__global__ __launch_bounds__(32) void wave_final(
    const int* __restrict__ target,
    const float* __restrict__ sr, const float* __restrict__ si,
    const float* __restrict__ acc, float* __restrict__ out) {
  const int lane = threadIdx.x;        // 0..31 == batch index

  int th = target[lane*2 + 0], tw = target[lane*2 + 1];
  size_t sidx = (size_t)lane*HW_ + (size_t)th*W_ + tw;
  float vr = sr[sidx], vi = si[sidx];
  float magv = sqrtf(vr*vr + vi*vi + 1e-8f);
  out[lane*9] = (THRESH_ - magv) * 12.f;   // none logit

  // B matrix 4x16: lanes0-15 col N hold rows K0(cos)/K1(sin); K2,K3 = 0
  const int n = lane & 15;
  float cth = 0.f, sth = 0.f;
  if (lane < 16 && n < 8) {
    float thk = 6.28318530717958647692f * (float)n / 8.f;
    cth = cosf(thk);  sth = sinf(thk);
  }
  v2f Bm = { cth, sth };
  // A 16x4: lanes0-15 row M hold K0(vr)/K1(vi); lanes16-31 K2,K3 = 0
  v2f A0 = { (lane < 16) ? vr : 0.f, (lane < 16) ? vi : 0.f };
  float vr1 = __shfl(vr, (lane & 15) + 16, 32);
  float vi1 = __shfl(vi, (lane & 15) + 16, 32);
  v2f A1 = { (lane < 16) ? vr1 : 0.f, (lane < 16) ? vi1 : 0.f };
  v8f C = {};
  v8f D0 = __builtin_amdgcn_wmma_f32_16x16x4_f32(false, A0, false, Bm, (short)0, C, false, false);
  v8f D1 = __builtin_amdgcn_wmma_f32_16x16x4_f32(false, A1, false, Bm, (short)0, C, false, false);

  // D layout: lane l holds col N=l%16, rows M = v + 8*(l/16) in VGPR v
  if (n < 8) {
    const int half = lane >> 4;
    for (int v = 0; v < 8; ++v) {
      int row = v + 8*half;
      out[row*9        + 1 + n] = (D0[v] - THRESH_) * 12.f;
      out[(16+row)*9   + 1 + n] = (D1[v] - THRESH_) * 12.f;
    }
  }

  if (lane == 0) {
    float normTot = 0.f;
    for (int s = 0; s < PSLOTS; ++s) normTot += acc[ACC_NORM0 + s];
    float delta = 0.f, pwTot = 0.f;
    for (int t = 1; t <= STEPS_; ++t) {
      const float* as = acc + ACC_STEP(t);
      float pw = 0.f, df = 0.f, nm = 0.f;
      for (int s = 0; s < PSLOTS; ++s) {
        pw += as[s];  df += as[PSLOTS + s];  nm += as[2*PSLOTS + s];
      }
      pwTot   += pw;
      delta   += sqrtf(df / (float)(2*NPIX));
      normTot += nm;
    }
    float sat = 0.f, lk = 0.f;
    for (int s = 0; s < PSLOTS; ++s) { sat += acc[ACC_SAT + s]; lk += acc[ACC_LEAK + s]; }
    // stats in jax pytree (alphabetical dict-key) order
    out[288] = delta / (float)STEPS_;                        // final_state_delta
    out[289] = normTot / (17.0f * (float)NPIX);              // phase_norm_by_step
    out[290] = sat / (float)(2*NPIX);                        // phase_saturation_rate
    out[291] = lk / (float)NPIX;                             // post_mask_wall_leak
    out[292] = pwTot / ((float)STEPS_ * (float)NPIX);        // pre_mask_wall_pressure
  }
}

// ---------------------------------------------------------------- launch
extern "C" void kernel_launch(void* const* d_in, const int* in_sizes, int n_in,
                              void* d_out, int out_size, void* d_ws, size_t ws_size,
                              hipStream_t stream) {
  (void)in_sizes; (void)n_in; (void)out_size; (void)ws_size;
  const float* x      = (const float*)d_in[0];
  const int*   target = (const int*)d_in[1];
  // d_in[2] = steps, fixed at 16 by the reference

  float* ws  = (float*)d_ws;
  float* sr  = ws;                 // state (updated in place)
  float* si  = sr  + NPIX;
  float* f0r = si  + NPIX;         // ping-pong flow buffers (stencil halo)
  float* f0i = f0r + NPIX;
  float* f1r = f0i + NPIX;
  float* f1i = f1r + NPIX;
  float* acc = f1i + NPIX;         // reduction partials

  hipMemsetAsync(acc, 0, ACC_TOTAL*sizeof(float), stream);

  dim3 blk(16, 16, 1), grd(W_/16, H_/16, B_);
  wave_init<<<grd, blk, 0, stream>>>(x, sr, si, f0r, f0i, acc);

  float* fr[2] = {f0r, f1r};
  float* fi[2] = {f0i, f1i};
  for (int t = 1; t <= STEPS_; ++t) {
    wave_step<<<grd, blk, 0, stream>>>(x, fr[(t-1)&1], fi[(t-1)&1],
                                       fr[t&1], fi[t&1], sr, si, acc, t);
  }
  wave_final<<<dim3(1), dim3(32), 0, stream>>>(target, sr, si, acc, (float*)d_out);
}